// gcnmask_81003083203455
// MI455X (gfx1250) — compile-verified
//
#include <hip/hip_runtime.h>

typedef __attribute__((ext_vector_type(2))) float v2f;
typedef __attribute__((ext_vector_type(8))) float v8f;

#define N_NODES 50000
#define DEG     16
#define F       64
#define NTILES  ((N_NODES + 15) / 16)   // 3125 (exact)
#define WPITCH  (F + 8)                 // LDS pitch for weight tiles (bank-conflict-free B frags)
#define XPITCH  (F + 4)                 // LDS pitch for x_new tiles (bank-conflict-free A frags)

__device__ __forceinline__ float sigmoidf(float t) {
    return 1.0f / (1.0f + __expf(-t));
}

// ---------------------------------------------------------------------------
// Kernel 1: xa = x @ wm_top ; xb = x @ wm_bot   (wm = weights_mask0, 128x64)
// One wave per 16-row tile of x; 4 waves per block.
// ---------------------------------------------------------------------------
__global__ __launch_bounds__(128)
void k1_dual_gemm(const float* __restrict__ x,
                  const float* __restrict__ wmask,   // 128 x 64 row-major
                  float* __restrict__ xa,
                  float* __restrict__ xb)
{
    __shared__ __align__(16) float lw[2 * F * WPITCH];   // top then bot, padded pitch

    const int tid = threadIdx.x;
    for (int i = tid; i < 2 * F * (F / 4); i += 128) {
        const int row = i >> 4;          // 0..127
        const int c4  = i & 15;          // 0..15
        const float4 v = *(const float4*)(wmask + row * F + c4 * 4);
        *(float4*)(&lw[row * WPITCH + c4 * 4]) = v;
    }
    __syncthreads();

    const int wave = tid >> 5;
    const int lane = tid & 31;
    const int l16  = lane & 15;
    const int hi   = lane >> 4;
    const int tile = blockIdx.x * 4 + wave;
    if (tile >= NTILES) return;          // wave-uniform: EXEC stays all-ones
    const int row0 = tile * 16;

    // A fragments: x tile 16x64, 16 k-steps of K=4
    v2f a[16];
    const float* xrow = x + (size_t)(row0 + l16) * F;
#pragma unroll
    for (int kk = 0; kk < 16; ++kk) {
        const int kb = kk * 4 + 2 * hi;
        a[kk].x = xrow[kb + 0];
        a[kk].y = xrow[kb + 1];
    }

#pragma unroll
    for (int nt = 0; nt < 4; ++nt) {
        const int n0 = nt * 16;
        v8f ca = {};
        v8f cb = {};
#pragma unroll
        for (int kk = 0; kk < 16; ++kk) {
            const int r = kk * 4 + 2 * hi;
            v2f bt, bb;
            bt.x = lw[(r + 0) * WPITCH + n0 + l16];
            bt.y = lw[(r + 1) * WPITCH + n0 + l16];
            bb.x = lw[(F + r + 0) * WPITCH + n0 + l16];
            bb.y = lw[(F + r + 1) * WPITCH + n0 + l16];
            ca = __builtin_amdgcn_wmma_f32_16x16x4_f32(false, a[kk], false, bt,
                                                       (short)0, ca, false, false);
            cb = __builtin_amdgcn_wmma_f32_16x16x4_f32(false, a[kk], false, bb,
                                                       (short)0, cb, false, false);
        }
#pragma unroll
        for (int r = 0; r < 8; ++r) {
            const int m = r + 8 * hi;
            xa[(size_t)(row0 + m) * F + n0 + l16] = ca[r];
            xb[(size_t)(row0 + m) * F + n0 + l16] = cb[r];
        }
    }
}

// ---------------------------------------------------------------------------
// Kernel 2: x_new[n] = x[n] + sum_d sigmoid(xa[n]+xb[j])*dm[n,d]*x[j];
//           support = x_new @ weight_0
// 8 waves per block cover 2 tiles (32 nodes): all 8 waves stream drop_mask
// (4 nodes each) for maximum memory-level parallelism, then waves 0-1 run
// the per-tile WMMA from LDS.
// ---------------------------------------------------------------------------
__global__ __launch_bounds__(256)
void k2_mask_gemm(const float* __restrict__ x,
                  const float* __restrict__ w0,      // 64x64
                  const float* __restrict__ xa,
                  const float* __restrict__ xb,
                  const float* __restrict__ dm,      // N x 16 x 64
                  const long long* __restrict__ nbr, // N x 16 (int64)
                  float* __restrict__ support)
{
    __shared__ __align__(16) float lw[F * WPITCH];
    __shared__ __align__(16) float lxn[2][16 * XPITCH];

    const int tid = threadIdx.x;
    for (int i = tid; i < F * (F / 4); i += 256) {
        const int row = i >> 4;
        const int c4  = i & 15;
        const float4 v = *(const float4*)(w0 + row * F + c4 * 4);
        *(float4*)(&lw[row * WPITCH + c4 * 4]) = v;
    }

    const int wave  = tid >> 5;
    const int lane  = tid & 31;
    const int l16   = lane & 15;
    const int hi    = lane >> 4;
    const int nbase = blockIdx.x * 32;       // first node of this block's 2 tiles

    // ---- Phase 1: streaming mask/aggregate. Each wave owns 4 nodes;
    //      lane owns features 2*lane, 2*lane+1.
    const int f0 = lane * 2;
    for (int mm = 0; mm < 4; ++mm) {
        const int li = wave * 4 + mm;        // 0..31 within block
        const int n  = nbase + li;
        if (n >= N_NODES) break;             // wave-uniform (no WMMA here)
        const float* xr0 = x  + (size_t)n * F;
        const float* xar = xa + (size_t)n * F;
        float acc0 = xr0[f0];
        float acc1 = xr0[f0 + 1];
        const float xan0 = xar[f0];
        const float xan1 = xar[f0 + 1];
        const long long* nb  = nbr + (size_t)n * DEG;
        const float*     dmr = dm  + (size_t)n * DEG * F;
        if (mm < 3 && n + 1 < N_NODES)
            __builtin_prefetch(dm + (size_t)(n + 1) * DEG * F, 0, 0);
#pragma unroll 8
        for (int d = 0; d < DEG; ++d) {
            const int j = (int)nb[d];
            const float* xbr = xb + (size_t)j * F;
            const float* xjr = x  + (size_t)j * F;
            const float s0 = sigmoidf(xan0 + xbr[f0]);
            const float s1 = sigmoidf(xan1 + xbr[f0 + 1]);
            acc0 += s0 * dmr[d * F + f0]     * xjr[f0];
            acc1 += s1 * dmr[d * F + f0 + 1] * xjr[f0 + 1];
        }
        lxn[li >> 4][(li & 15) * XPITCH + f0]     = acc0;
        lxn[li >> 4][(li & 15) * XPITCH + f0 + 1] = acc1;
    }
    __syncthreads();

    // ---- Phase 2: waves 0-1 each compute one tile's support = x_new @ w0.
    if (wave >= 2) return;                   // wave-uniform
    const int tile = blockIdx.x * 2 + wave;
    if (tile >= NTILES) return;              // wave-uniform
    const int row0 = tile * 16;

    v2f a[16];
#pragma unroll
    for (int kk = 0; kk < 16; ++kk) {
        const int kb = kk * 4 + 2 * hi;
        a[kk].x = lxn[wave][l16 * XPITCH + kb + 0];
        a[kk].y = lxn[wave][l16 * XPITCH + kb + 1];
    }
#pragma unroll
    for (int nt = 0; nt < 4; ++nt) {
        const int n0 = nt * 16;
        v8f c = {};
#pragma unroll
        for (int kk = 0; kk < 16; ++kk) {
            const int r = kk * 4 + 2 * hi;
            v2f b;
            b.x = lw[(r + 0) * WPITCH + n0 + l16];
            b.y = lw[(r + 1) * WPITCH + n0 + l16];
            c = __builtin_amdgcn_wmma_f32_16x16x4_f32(false, a[kk], false, b,
                                                      (short)0, c, false, false);
        }
#pragma unroll
        for (int r = 0; r < 8; ++r) {
            const int m = r + 8 * hi;
            support[(size_t)(row0 + m) * F + n0 + l16] = c[r];
        }
    }
}

// ---------------------------------------------------------------------------
// Kernel 3: out[n] = bias + sum_d adj[n,d] * support[nbr[n,d]]
// One wave per node; lane owns 2 features.
// ---------------------------------------------------------------------------
__global__ __launch_bounds__(256)
void k3_aggregate(const float* __restrict__ support,
                  const float* __restrict__ adj,     // N x 16
                  const long long* __restrict__ nbr, // N x 16
                  const float* __restrict__ bias,    // 64
                  float* __restrict__ out)
{
    const int tid  = threadIdx.x;
    const int wave = tid >> 5;
    const int lane = tid & 31;
    const int n = blockIdx.x * 8 + wave;
    if (n >= N_NODES) return;
    const int f0 = lane * 2;

    float acc0 = bias[f0];
    float acc1 = bias[f0 + 1];
    const long long* nb = nbr + (size_t)n * DEG;
    const float*     ar = adj + (size_t)n * DEG;
#pragma unroll 4
    for (int d = 0; d < DEG; ++d) {
        const int j = (int)nb[d];
        const float w = ar[d];
        const float* sr = support + (size_t)j * F;
        acc0 += w * sr[f0];
        acc1 += w * sr[f0 + 1];
    }
    out[(size_t)n * F + f0]     = acc0;
    out[(size_t)n * F + f0 + 1] = acc1;
}

// ---------------------------------------------------------------------------
extern "C" void kernel_launch(void* const* d_in, const int* in_sizes, int n_in,
                              void* d_out, int out_size, void* d_ws, size_t ws_size,
                              hipStream_t stream) {
    const float*     x    = (const float*)d_in[0];
    const float*     w0   = (const float*)d_in[1];
    const float*     wm   = (const float*)d_in[2];   // 128x64: top rows then bot rows
    const float*     bias = (const float*)d_in[3];
    const float*     adj  = (const float*)d_in[4];
    const float*     dm   = (const float*)d_in[5];
    const long long* nbr  = (const long long*)d_in[6];
    float* out = (float*)d_out;

    float* xa      = (float*)d_ws;
    float* xb      = xa + (size_t)N_NODES * F;
    float* support = xb + (size_t)N_NODES * F;

    k1_dual_gemm<<<(NTILES + 3) / 4, 128, 0, stream>>>(x, wm, xa, xb);
    k2_mask_gemm<<<(NTILES + 1) / 2, 256, 0, stream>>>(x, w0, xa, xb, dm, nbr, support);
    k3_aggregate<<<(N_NODES + 7) / 8, 256, 0, stream>>>(support, adj, nbr, bias, out);
}